// TTLinear_38551626449009
// MI455X (gfx1250) — compile-verified
//
#include <hip/hip_runtime.h>
#include <hip/hip_bf16.h>

// ---------------------------------------------------------------------------
// TTT scan for MI455X (gfx1250): f32 WMMA GEMMs (V_WMMA_F32_16X16X4_F32).
// One templated GEMM kernel covers NT / NN / TN operand layouts with fused
// epilogues. Sequential scan expressed as a per-step launch chain on `stream`.
// ---------------------------------------------------------------------------

typedef float v2f __attribute__((ext_vector_type(2)));
typedef float v8f __attribute__((ext_vector_type(8)));

__device__ __forceinline__ float gelu_f(float x) {
    return 0.5f * x * (1.0f + erff(x * 0.70710678118654752440f));
}
__device__ __forceinline__ float dgelu_f(float x) {
    float cdf = 0.5f * (1.0f + erff(x * 0.70710678118654752440f));
    float pdf = 0.39894228040143267794f * expf(-0.5f * x * x);
    return cdf + x * pdf;
}

// Epilogue codes
// 0: OUT = acc
// 1: p = acc + bias[n]; AUX = p; OUT = gelu(p)
// 2: OUT = resid + acc + bias[n]
// 3: OUT = scale * (resid + acc + bias[n] - Y)
// 4: OUT = acc * dgelu(premat)
// 5: OUT -= scale * acc        (in-place SGD on the weight matrix)
template <int EPI>
__device__ __forceinline__ void epi_store(float* OUT, float* AUX,
                                          const float* bias, const float* resid,
                                          const float* Ymat, const float* premat,
                                          int N, int gm, int gn, float acc,
                                          float scale) {
    size_t idx = (size_t)gm * N + gn;
    if (EPI == 0) {
        OUT[idx] = acc;
    } else if (EPI == 1) {
        float p = acc + bias[gn];
        AUX[idx] = p;
        OUT[idx] = gelu_f(p);
    } else if (EPI == 2) {
        OUT[idx] = resid[idx] + acc + bias[gn];
    } else if (EPI == 3) {
        OUT[idx] = scale * (resid[idx] + acc + bias[gn] - Ymat[idx]);
    } else if (EPI == 4) {
        OUT[idx] = acc * dgelu_f(premat[idx]);
    } else {
        OUT[idx] -= scale * acc;
    }
}

// C[M,N] = opA(A) @ opB(B), f32 WMMA 16x16x4.
// OPA==0: A stored row-major [M,K].      OPA==1: A stored [K,M] (computes A^T @ ...).
// OPB==0: B stored [N,K] (dot of rows).  OPB==1: B stored [K,N].
// M, N, K must be multiples of 64/64/16 (true for every shape here).
template <int OPA, int OPB, int EPI>
__global__ __launch_bounds__(256) void gemm_wmma_f32(
    const float* __restrict__ A, const float* __restrict__ B,
    float* __restrict__ OUT, float* __restrict__ AUX,
    const float* __restrict__ bias, const float* __restrict__ resid,
    const float* __restrict__ Ymat, const float* __restrict__ premat,
    int M, int N, int K, float scale) {
    constexpr int BM = 64, BN = 64, BK = 16;
    constexpr int BNP = BN + 16;  // pad so lanes16-31 hit a disjoint bank group
    __shared__ v2f As2[BM][BK / 2 + 1];  // element [m][k/2] = (A(m,k),A(m,k+1))
    __shared__ v2f Bs2[BK / 2][BNP];     // element [k/2][n] = (B(k,n),B(k+1,n))

    const int tid = threadIdx.x;
    const int lane = tid & 31;
    const int wid = tid >> 5;
    const int wm = (wid >> 1) * 16;  // wave row offset in block tile (4 waves in M)
    const int wn = (wid & 1) * 32;   // wave col offset (2 waves in N, 2 tiles each)
    const int m0 = blockIdx.y * BM;
    const int n0 = blockIdx.x * BN;

    v8f acc0 = {0.f, 0.f, 0.f, 0.f, 0.f, 0.f, 0.f, 0.f};
    v8f acc1 = {0.f, 0.f, 0.f, 0.f, 0.f, 0.f, 0.f, 0.f};

    float* asf = reinterpret_cast<float*>(&As2[0][0]);
    float* bsf = reinterpret_cast<float*>(&Bs2[0][0]);
    constexpr int ASF = (BK / 2 + 1) * 2;  // floats per As2 row (18)
    constexpr int BSF = BNP * 2;           // floats per Bs2 row (160)

    for (int k0 = 0; k0 < K; k0 += BK) {
        // ---- stage A tile ----
        if (OPA == 0) {
            int r = tid >> 2, seg = tid & 3;
            const float4 v = *reinterpret_cast<const float4*>(
                A + (size_t)(m0 + r) * K + k0 + seg * 4);
            v2f lo; lo.x = v.x; lo.y = v.y;
            v2f hi; hi.x = v.z; hi.y = v.w;
            As2[r][seg * 2 + 0] = lo;
            As2[r][seg * 2 + 1] = hi;
        } else {
            // A global is [K][M]; transpose into As2 (As element (m,k) = G[k][m])
            int kk = tid >> 4, seg = tid & 15;
            const float4 v = *reinterpret_cast<const float4*>(
                A + (size_t)(k0 + kk) * M + m0 + seg * 4);
            asf[(seg * 4 + 0) * ASF + kk] = v.x;
            asf[(seg * 4 + 1) * ASF + kk] = v.y;
            asf[(seg * 4 + 2) * ASF + kk] = v.z;
            asf[(seg * 4 + 3) * ASF + kk] = v.w;
        }
        // ---- stage B tile ----  (Bs element (k,n) at bsf[(k>>1)*BSF + n*2 + (k&1)])
        if (OPB == 0) {
            int n = tid >> 2, seg = tid & 3;
            const float4 v = *reinterpret_cast<const float4*>(
                B + (size_t)(n0 + n) * K + k0 + seg * 4);
            bsf[(seg * 2 + 0) * BSF + n * 2 + 0] = v.x;
            bsf[(seg * 2 + 0) * BSF + n * 2 + 1] = v.y;
            bsf[(seg * 2 + 1) * BSF + n * 2 + 0] = v.z;
            bsf[(seg * 2 + 1) * BSF + n * 2 + 1] = v.w;
        } else {
            int kk = tid >> 4, seg = tid & 15;
            const float4 v = *reinterpret_cast<const float4*>(
                B + (size_t)(k0 + kk) * N + n0 + seg * 4);
            int kr = (kk >> 1) * BSF + (kk & 1);
            bsf[kr + (seg * 4 + 0) * 2] = v.x;
            bsf[kr + (seg * 4 + 1) * 2] = v.y;
            bsf[kr + (seg * 4 + 2) * 2] = v.z;
            bsf[kr + (seg * 4 + 3) * 2] = v.w;
        }
        __syncthreads();

        // ---- WMMA inner loop over K chunks of 4 ----
        const int hi = lane >> 4;    // 0: K=k,k+1 ; 1: K=k+2,k+3  (ISA f32 layout)
        const int l15 = lane & 15;
#pragma unroll
        for (int kc = 0; kc < BK; kc += 4) {
            const int k2 = (kc >> 1) + hi;
            v2f a = As2[wm + l15][k2];
            v2f b0 = Bs2[k2][wn + l15];
            v2f b1 = Bs2[k2][wn + 16 + l15];
            acc0 = __builtin_amdgcn_wmma_f32_16x16x4_f32(
                false, a, false, b0, (short)0, acc0, false, false);
            acc1 = __builtin_amdgcn_wmma_f32_16x16x4_f32(
                false, a, false, b1, (short)0, acc1, false, false);
        }
        __syncthreads();
    }

    // ---- epilogue: C layout = VGPR r -> M=r (lanes0-15) / M=r+8 (lanes16-31) ----
    const int gm0 = m0 + wm + (lane >> 4) * 8;
    const int gn = n0 + wn + (lane & 15);
#pragma unroll
    for (int r = 0; r < 8; ++r) {
        epi_store<EPI>(OUT, AUX, bias, resid, Ymat, premat, N, gm0 + r, gn,
                       acc0[r], scale);
        epi_store<EPI>(OUT, AUX, bias, resid, Ymat, premat, N, gm0 + r, gn + 16,
                       acc1[r], scale);
    }
}

// b[j] -= lr * sum_m G[m][j]   (deterministic serial column reduction)
__global__ __launch_bounds__(256) void bias_colsum_sgd(
    float* __restrict__ bparam, const float* __restrict__ G, int Mrows,
    int Ncols, float lr) {
    int j = blockIdx.x * blockDim.x + threadIdx.x;
    if (j < Ncols) {
        float s = 0.f;
        for (int m = 0; m < Mrows; ++m) s += G[(size_t)m * Ncols + j];
        bparam[j] -= lr * s;
    }
}

extern "C" void kernel_launch(void* const* d_in, const int* in_sizes, int n_in,
                              void* d_out, int out_size, void* d_ws,
                              size_t ws_size, hipStream_t stream) {
    (void)in_sizes; (void)n_in; (void)out_size;
    constexpr int T = 128, Nb = 256, D = 1024, H = 1024, H4 = 256;
    constexpr float LR = 1e-3f;
    const float escale = 2.0f / (float)((size_t)Nb * H);

    const float* in_seq = (const float*)d_in[0];
    const float* t_k = (const float*)d_in[1];
    const float* t_v = (const float*)d_in[2];
    const float* t_q = (const float*)d_in[3];
    float* out = (float*)d_out;

    // ---- workspace layout (floats) ----
    float* ws = (float*)d_ws;
    size_t off = 0;
    float* W1c = ws + off; off += (size_t)H4 * H;
    float* W2c = ws + off; off += (size_t)H * H4;
    float* b1c = ws + off; off += H4;
    float* b2c = ws + off; off += H;
    float* pre1 = ws + off; off += (size_t)Nb * H4;   // pre-act (reused for S6a)
    float* hbuf = ws + off; off += (size_t)Nb * H4;   // gelu(pre1)
    float* dpre = ws + off; off += (size_t)Nb * H4;   // grad at pre1
    float* hq   = ws + off; off += (size_t)Nb * H4;   // gelu on query path
    float* Ebuf = ws + off; off += (size_t)Nb * H;    // scaled error
    const size_t perT = (size_t)Nb * H;
    // Big path: precompute all T projections in parallel (403 MB); else per-step.
    const bool big = ws_size >= (off + 3 * (size_t)T * perT) * sizeof(float);
    float *Zb = ws + off, *Yb, *Qb;
    if (big) { off += (size_t)T * perT; Yb = ws + off; off += (size_t)T * perT;
               Qb = ws + off; }
    else     { off += perT; Yb = ws + off; off += perT; Qb = ws + off; }

    // mutable copies of the fast-weights (inputs must stay untouched)
    hipMemcpyAsync(W1c, d_in[4], sizeof(float) * (size_t)H4 * H,
                   hipMemcpyDeviceToDevice, stream);
    hipMemcpyAsync(b1c, d_in[5], sizeof(float) * H4, hipMemcpyDeviceToDevice, stream);
    hipMemcpyAsync(W2c, d_in[6], sizeof(float) * (size_t)H * H4,
                   hipMemcpyDeviceToDevice, stream);
    hipMemcpyAsync(b2c, d_in[7], sizeof(float) * H, hipMemcpyDeviceToDevice, stream);

    dim3 blk(256);
    auto g = [](int Mm, int Nn) { return dim3(Nn / 64, Mm / 64); };
    const float* NPF = nullptr;
    float* NPM = nullptr;

    if (big) {  // hoisted, fully parallel projections: [T*N, H] = [T*N, D] @ W^T
        gemm_wmma_f32<0, 0, 0><<<g(T * Nb, H), blk, 0, stream>>>(
            in_seq, t_k, Zb, NPM, NPF, NPF, NPF, NPF, T * Nb, H, D, 0.f);
        gemm_wmma_f32<0, 0, 0><<<g(T * Nb, H), blk, 0, stream>>>(
            in_seq, t_v, Yb, NPM, NPF, NPF, NPF, NPF, T * Nb, H, D, 0.f);
        gemm_wmma_f32<0, 0, 0><<<g(T * Nb, H), blk, 0, stream>>>(
            in_seq, t_q, Qb, NPM, NPF, NPF, NPF, NPF, T * Nb, H, D, 0.f);
    }

    for (int t = 0; t < T; ++t) {
        const float* x = in_seq + (size_t)t * Nb * D;
        float *Zt, *Yt, *Qt;
        if (big) { Zt = Zb + (size_t)t * perT; Yt = Yb + (size_t)t * perT;
                   Qt = Qb + (size_t)t * perT; }
        else {
            Zt = Zb; Yt = Yb; Qt = Qb;
            gemm_wmma_f32<0, 0, 0><<<g(Nb, H), blk, 0, stream>>>(
                x, t_k, Zt, NPM, NPF, NPF, NPF, NPF, Nb, H, D, 0.f);
            gemm_wmma_f32<0, 0, 0><<<g(Nb, H), blk, 0, stream>>>(
                x, t_v, Yt, NPM, NPF, NPF, NPF, NPF, Nb, H, D, 0.f);
            gemm_wmma_f32<0, 0, 0><<<g(Nb, H), blk, 0, stream>>>(
                x, t_q, Qt, NPM, NPF, NPF, NPF, NPF, Nb, H, D, 0.f);
        }
        // S1: pre1 = Zt@W1^T + b1 ; h = gelu(pre1)
        gemm_wmma_f32<0, 0, 1><<<g(Nb, H4), blk, 0, stream>>>(
            Zt, W1c, hbuf, pre1, b1c, NPF, NPF, NPF, Nb, H4, H, 0.f);
        // S2: E = (2/NH) * (Zt + h@W2^T + b2 - Yt)
        gemm_wmma_f32<0, 0, 3><<<g(Nb, H), blk, 0, stream>>>(
            hbuf, W2c, Ebuf, NPM, b2c, Zt, Yt, NPF, Nb, H, H4, escale);
        // S4 (before W2 update!): dpre = (E @ W2) * gelu'(pre1)
        gemm_wmma_f32<0, 1, 4><<<g(Nb, H4), blk, 0, stream>>>(
            Ebuf, W2c, dpre, NPM, NPF, NPF, NPF, pre1, Nb, H4, H, 0.f);
        // S3: W2 -= LR * (E^T @ h)      [H,H4] = [Nb,H]^T @ [Nb,H4]
        gemm_wmma_f32<1, 1, 5><<<g(H, H4), blk, 0, stream>>>(
            Ebuf, hbuf, W2c, NPM, NPF, NPF, NPF, NPF, H, H4, Nb, LR);
        // b2 -= LR * colsum(E)
        bias_colsum_sgd<<<dim3(H / 256), blk, 0, stream>>>(b2c, Ebuf, Nb, H, LR);
        // S5: W1 -= LR * (dpre^T @ Zt)  [H4,H] = [Nb,H4]^T @ [Nb,H]
        gemm_wmma_f32<1, 1, 5><<<g(H4, H), blk, 0, stream>>>(
            dpre, Zt, W1c, NPM, NPF, NPF, NPF, NPF, H4, H, Nb, LR);
        // b1 -= LR * colsum(dpre)
        bias_colsum_sgd<<<dim3(H4 / 256), blk, 0, stream>>>(b1c, dpre, Nb, H4, LR);
        // S6a: hq = gelu(Qt@W1^T + b1)  (pre1 buffer reused as scratch)
        gemm_wmma_f32<0, 0, 1><<<g(Nb, H4), blk, 0, stream>>>(
            Qt, W1c, hq, pre1, b1c, NPF, NPF, NPF, Nb, H4, H, 0.f);
        // S6b: out_t = Qt + hq@W2^T + b2
        gemm_wmma_f32<0, 0, 2><<<g(Nb, H), blk, 0, stream>>>(
            hq, W2c, out + (size_t)t * perT, NPM, b2c, Qt, NPF, NPF, Nb, H, H4,
            0.f);
    }
}